// CriticGNN_40767829574201
// MI455X (gfx1250) — compile-verified
//
#include <hip/hip_runtime.h>
#include <math.h>

#define N_NODES  50000
#define N_EDGES  800000
#define NODE_DIM 32
#define EDGE_DIM 8
#define TEMP_HID 32
#define M_COMP   10000
#define T_STEPS  20
#define HID      64
#define NEG_SLOPE 0.2f
#define E2       (N_EDGES + N_NODES)
#define IN_FEAT  (NODE_DIM + TEMP_HID + 8)   // 72

typedef __attribute__((ext_vector_type(2))) float v2f;
typedef __attribute__((ext_vector_type(8))) float v8f;

// ---------- helpers ----------
__device__ __forceinline__ unsigned fkey(float x) {
  unsigned b = __float_as_uint(x);
  return (b & 0x80000000u) ? ~b : (b | 0x80000000u);
}
__device__ __forceinline__ float funkey(unsigned k) {
  unsigned b = (k & 0x80000000u) ? (k & 0x7fffffffu) : ~k;
  return __uint_as_float(b);
}

__global__ void zero4_f32(float4* __restrict__ p, long long n4) {
  long long i = (long long)blockIdx.x * blockDim.x + threadIdx.x;
  if (i < n4) p[i] = make_float4(0.f, 0.f, 0.f, 0.f);
}

// ---------- GEMM via V_WMMA_F32_16X16X4_F32: C[MxN] = act(A[MxK] @ W[NxK]^T + bias) ----------
// REQUIREMENT: M % 16 == 0 and N % 16 == 0 (true for all launches: M=50000, N in {64,128}).
template <int K, int ACT, bool BIAS>
__global__ void gemm_wt(const float* __restrict__ A, const float* __restrict__ W,
                        const float* __restrict__ bias, float* __restrict__ C,
                        int M, int N) {
  const int wid = blockIdx.x * (blockDim.x >> 5) + (threadIdx.x >> 5);
  const int tN = N >> 4;
  const int tM = M >> 4;
  if (wid >= tM * tN) return;               // wave-uniform exit; EXEC stays all-ones
  const int tm   = (wid / tN) << 4;
  const int tn   = (wid % tN) << 4;
  const int lane = threadIdx.x & 31;
  const int half = lane >> 4;               // K split across half-waves
  const int l16  = lane & 15;
  const float* Ap = A + (size_t)(tm + l16) * K + 2 * half;
  const float* Wp = W + (size_t)(tn + l16) * K + 2 * half;
  v8f acc = {0.f, 0.f, 0.f, 0.f, 0.f, 0.f, 0.f, 0.f};
#pragma unroll
  for (int kb = 0; kb < K; kb += 4) {
    // lanes 0-15 carry K=kb,kb+1 ; lanes 16-31 carry K=kb+2,kb+3 (A layout; B symmetric)
    const v2f a = *(const v2f*)(Ap + kb);   // 8B-aligned -> global_load_b64
    const v2f b = *(const v2f*)(Wp + kb);
    acc = __builtin_amdgcn_wmma_f32_16x16x4_f32(false, a, false, b, (short)0, acc,
                                                false, false);
  }
  const int n = tn + l16;
  float* Cp = C + (size_t)(tm + 8 * half) * N + n;
#pragma unroll
  for (int r = 0; r < 8; ++r) {             // C/D layout: VGPR r -> M=tm+r (+8 for upper half)
    float v = acc[r];
    if (BIAS) v += bias[n];
    if (ACT == 1) v = v > 0.f ? v : 0.f;    // relu
    Cp[(size_t)r * N] = v;
  }
}

// ---------- GRU: one wave per component, lane = hidden unit (H == 32 == wave32) ----------
__global__ void gru_kernel(const float* __restrict__ seq, const float* __restrict__ Wih,
                           const float* __restrict__ Whh, const float* __restrict__ bih,
                           const float* __restrict__ bhh, float* __restrict__ Ht) {
  __shared__ float sW[3 * 32 * 32];         // transposed per gate: sW[g*1024 + k*32 + l]
  for (int i = threadIdx.x; i < 3 * 32 * 32; i += blockDim.x) {
    int g = i >> 10, rem = i & 1023, k = rem >> 5, l = rem & 31;
    sW[i] = Whh[(g * 32 + l) * 32 + k];
  }
  __syncthreads();
  const int b = blockIdx.x * (blockDim.x >> 5) + (threadIdx.x >> 5);
  const int l = threadIdx.x & 31;
  if (b >= M_COMP) return;
  const float wr0 = Wih[l * 2],        wr1 = Wih[l * 2 + 1];
  const float wz0 = Wih[(32 + l) * 2], wz1 = Wih[(32 + l) * 2 + 1];
  const float wn0 = Wih[(64 + l) * 2], wn1 = Wih[(64 + l) * 2 + 1];
  const float bir = bih[l], biz = bih[32 + l], bin_ = bih[64 + l];
  const float bhr = bhh[l], bhz = bhh[32 + l], bhn = bhh[64 + l];
  float h = 0.f;
  const float* sp = seq + (size_t)b * T_STEPS * 2;
  for (int t = 0; t < T_STEPS; ++t) {
    const float x0 = sp[t * 2], x1 = sp[t * 2 + 1];
    const float gir = wr0 * x0 + wr1 * x1 + bir;
    const float giz = wz0 * x0 + wz1 * x1 + biz;
    const float gin = wn0 * x0 + wn1 * x1 + bin_;
    float ghr = bhr, ghz = bhz, ghn = bhn;
    for (int k = 0; k < 32; ++k) {
      const float hk = __shfl(h, k, 32);    // wave-uniform k -> v_readlane broadcast
      ghr += sW[k * 32 + l] * hk;
      ghz += sW[1024 + k * 32 + l] * hk;
      ghn += sW[2048 + k * 32 + l] * hk;
    }
    const float r = 1.f / (1.f + __expf(-(gir + ghr)));
    const float z = 1.f / (1.f + __expf(-(giz + ghz)));
    const float n = tanhf(gin + r * ghn);
    h = (1.f - z) * n + z * h;
  }
  Ht[(size_t)b * 32 + l] = h;
}

// ---------- node input features ----------
__global__ void fill_infeat_kernel(const float* __restrict__ x, const int* __restrict__ ntype,
                                   const float* __restrict__ temb, float* __restrict__ infeat) {
  long long i = (long long)blockIdx.x * blockDim.x + threadIdx.x;
  if (i >= (long long)N_NODES * IN_FEAT) return;
  const int n = (int)(i / IN_FEAT), c = (int)(i % IN_FEAT);
  float v;
  if (c < NODE_DIM)                 v = x[(size_t)n * NODE_DIM + c];
  else if (c < NODE_DIM + TEMP_HID) v = 0.f;
  else                              v = temb[ntype[n] * 8 + (c - NODE_DIM - TEMP_HID)];
  infeat[i] = v;
}

__global__ void scatter_ht_kernel(const int* __restrict__ comp_idx, const float* __restrict__ Ht,
                                  float* __restrict__ infeat) {
  const int i = blockIdx.x * blockDim.x + threadIdx.x;
  if (i >= M_COMP * TEMP_HID) return;
  const int ci = i >> 5, c = i & 31;
  infeat[(size_t)comp_idx[ci] * IN_FEAT + NODE_DIM + c] = Ht[i];
}

// ---------- self-loop edge_attr = mean incoming edge_attr ----------
__global__ void deg_easum_kernel(const int* __restrict__ eidx, const float* __restrict__ ea,
                                 float* __restrict__ deg, float* __restrict__ easum) {
  const int e = blockIdx.x * blockDim.x + threadIdx.x;
  if (e >= N_EDGES) return;
  const int dst = eidx[N_EDGES + e];
  atomicAdd(&deg[dst], 1.f);
#pragma unroll
  for (int j = 0; j < 8; ++j) atomicAdd(&easum[(size_t)dst * 8 + j], ea[(size_t)e * 8 + j]);
}

__global__ void ealoop_div_kernel(const float* __restrict__ deg, float* __restrict__ ealoop) {
  const int i = blockIdx.x * blockDim.x + threadIdx.x;
  if (i >= N_NODES * 8) return;
  float d = deg[i >> 3];
  d = d > 1.f ? d : 1.f;
  ealoop[i] = ealoop[i] / d;
}

// ---------- attention scores (one wave per edge, lane = 4 of 128 head-channels) ----------
__global__ void edge_score_kernel(const int* __restrict__ eidx, const float* __restrict__ ea,
                                  const float* __restrict__ xl, const float* __restrict__ xr,
                                  const float* __restrict__ We, const float* __restrict__ att,
                                  float* __restrict__ score, unsigned* __restrict__ smax) {
  __shared__ float sWe[128 * 8];
  __shared__ float sAtt[128];
  for (int i = threadIdx.x; i < 128 * 8; i += blockDim.x) sWe[i] = We[i];
  for (int i = threadIdx.x; i < 128; i += blockDim.x) sAtt[i] = att[i];
  __syncthreads();
  const int e = blockIdx.x * (blockDim.x >> 5) + (threadIdx.x >> 5);
  if (e >= N_EDGES) return;
  const int lane = threadIdx.x & 31;
  const int src = eidx[e];
  const int dst = eidx[N_EDGES + e];
  const float4 ea0 = *(const float4*)(ea + (size_t)e * 8);
  const float4 ea1 = *(const float4*)(ea + (size_t)e * 8 + 4);
  const int c0 = lane * 4;                  // head = c0>>6 -> lanes 0-15 head0, 16-31 head1
  const float4 xls = *(const float4*)(xl + (size_t)src * 128 + c0);
  const float4 xrd = *(const float4*)(xr + (size_t)dst * 128 + c0);
  const float xlv[4] = {xls.x, xls.y, xls.z, xls.w};
  const float xrv[4] = {xrd.x, xrd.y, xrd.z, xrd.w};
  float s = 0.f;
#pragma unroll
  for (int j = 0; j < 4; ++j) {
    const int c = c0 + j;
    const float* wrow = &sWe[c * 8];
    float ev = ea0.x * wrow[0] + ea0.y * wrow[1] + ea0.z * wrow[2] + ea0.w * wrow[3] +
               ea1.x * wrow[4] + ea1.y * wrow[5] + ea1.z * wrow[6] + ea1.w * wrow[7];
    float m = xlv[j] + xrv[j] + ev;
    m = m > 0.f ? m : NEG_SLOPE * m;
    s += m * sAtt[c];
  }
#pragma unroll
  for (int off = 1; off < 16; off <<= 1) s += __shfl_xor(s, off, 32);  // per-half reduce
  if ((lane & 15) == 0) {
    const int hd = lane >> 4;
    score[(size_t)e * 2 + hd] = s;
    atomicMax(&smax[(size_t)dst * 2 + hd], fkey(s));
  }
}

__global__ void loop_score_kernel(const float* __restrict__ ealoop, const float* __restrict__ xl,
                                  const float* __restrict__ xr, const float* __restrict__ We,
                                  const float* __restrict__ att, float* __restrict__ score,
                                  unsigned* __restrict__ smax) {
  __shared__ float sWe[128 * 8];
  __shared__ float sAtt[128];
  for (int i = threadIdx.x; i < 128 * 8; i += blockDim.x) sWe[i] = We[i];
  for (int i = threadIdx.x; i < 128; i += blockDim.x) sAtt[i] = att[i];
  __syncthreads();
  const int n = blockIdx.x * (blockDim.x >> 5) + (threadIdx.x >> 5);
  if (n >= N_NODES) return;
  const int lane = threadIdx.x & 31;
  const float4 ea0 = *(const float4*)(ealoop + (size_t)n * 8);
  const float4 ea1 = *(const float4*)(ealoop + (size_t)n * 8 + 4);
  const int c0 = lane * 4;
  const float4 xls = *(const float4*)(xl + (size_t)n * 128 + c0);
  const float4 xrd = *(const float4*)(xr + (size_t)n * 128 + c0);
  const float xlv[4] = {xls.x, xls.y, xls.z, xls.w};
  const float xrv[4] = {xrd.x, xrd.y, xrd.z, xrd.w};
  float s = 0.f;
#pragma unroll
  for (int j = 0; j < 4; ++j) {
    const int c = c0 + j;
    const float* wrow = &sWe[c * 8];
    float ev = ea0.x * wrow[0] + ea0.y * wrow[1] + ea0.z * wrow[2] + ea0.w * wrow[3] +
               ea1.x * wrow[4] + ea1.y * wrow[5] + ea1.z * wrow[6] + ea1.w * wrow[7];
    float m = xlv[j] + xrv[j] + ev;
    m = m > 0.f ? m : NEG_SLOPE * m;
    s += m * sAtt[c];
  }
#pragma unroll
  for (int off = 1; off < 16; off <<= 1) s += __shfl_xor(s, off, 32);
  if ((lane & 15) == 0) {
    const int hd = lane >> 4;
    score[(size_t)(N_EDGES + n) * 2 + hd] = s;
    atomicMax(&smax[(size_t)n * 2 + hd], fkey(s));
  }
}

__global__ void expdenom_kernel(const int* __restrict__ eidx, float* __restrict__ score,
                                const unsigned* __restrict__ smax, float* __restrict__ denom) {
  const long long i = (long long)blockIdx.x * blockDim.x + threadIdx.x;
  if (i >= E2) return;
  const int dst = (i < N_EDGES) ? eidx[N_EDGES + i] : (int)(i - N_EDGES);
#pragma unroll
  for (int h = 0; h < 2; ++h) {
    const float m = funkey(smax[(size_t)dst * 2 + h]);
    const float p = __expf(score[i * 2 + h] - m);
    score[i * 2 + h] = p;
    atomicAdd(&denom[(size_t)dst * 2 + h], p);
  }
}

__global__ void aggregate_kernel(const int* __restrict__ eidx, const float* __restrict__ score,
                                 const float* __restrict__ denom, const float* __restrict__ xl,
                                 float* __restrict__ aggr) {
  const long long i = (long long)blockIdx.x * (blockDim.x >> 5) + (threadIdx.x >> 5);
  if (i >= E2) return;
  const int lane = threadIdx.x & 31;
  int src, dst;
  if (i < N_EDGES) { src = eidx[i]; dst = eidx[N_EDGES + i]; }
  else             { src = dst = (int)(i - N_EDGES); }
  const int hd = lane >> 4;
  const float alpha = score[i * 2 + hd] / denom[(size_t)dst * 2 + hd];
  const int c0 = lane * 4;
  const float4 v = *(const float4*)(xl + (size_t)src * 128 + c0);
  float* o = aggr + (size_t)dst * 128 + c0;
  atomicAdd(o + 0, alpha * v.x);
  atomicAdd(o + 1, alpha * v.y);
  atomicAdd(o + 2, alpha * v.z);
  atomicAdd(o + 3, alpha * v.w);
}

__global__ void finalize_kernel(const float* __restrict__ aggr, const float* __restrict__ bias,
                                float* __restrict__ hout, int act) {
  const int i = blockIdx.x * blockDim.x + threadIdx.x;
  if (i >= N_NODES * HID) return;
  const int n = i >> 6, c = i & 63;
  float v = 0.5f * (aggr[(size_t)n * 128 + c] + aggr[(size_t)n * 128 + 64 + c]) + bias[c];
  if (act == 1) v = v > 0.f ? v : (__expf(v) - 1.f);   // elu
  hout[i] = v;
}

// ---------- global reductions + heads ----------
__global__ void sumH_kernel(const float* __restrict__ H, float* __restrict__ red) {
  const int c = threadIdx.x;                 // blockDim 64
  const int r0 = blockIdx.x * 128;
  const int r1 = min(r0 + 128, N_NODES);
  float s = 0.f;
  for (int r = r0; r < r1; ++r) s += H[(size_t)r * 64 + c];
  atomicAdd(&red[c], s);
}

__global__ void sumHc_kernel(const float* __restrict__ H, const int* __restrict__ comp_idx,
                             float* __restrict__ red) {
  const int c = threadIdx.x;
  const int r0 = blockIdx.x * 128;
  const int r1 = min(r0 + 128, M_COMP);
  float s = 0.f;
  for (int i = r0; i < r1; ++i) s += H[(size_t)comp_idx[i] * 64 + c];
  atomicAdd(&red[64 + c], s);
}

__global__ void final_kernel(const float* __restrict__ red, const float* __restrict__ gW,
                             const float* __restrict__ gb, const float* __restrict__ v1W,
                             const float* __restrict__ v1b, const float* __restrict__ v2W,
                             const float* __restrict__ v2b, float* __restrict__ out) {
  __shared__ float hm[64];
  __shared__ float ft[128];
  __shared__ float part[4];
  const int t = threadIdx.x;                 // blockDim 128
  if (t < 64) {
    hm[t] = red[t] / (float)N_NODES;
    ft[t] = red[64 + t] / (float)M_COMP;     // Hc in ft[0..63]
  }
  __syncthreads();
  if (t < 64) {
    float s = gb[t];
    for (int k = 0; k < 64; ++k) s += hm[k] * gW[t * 64 + k];
    ft[64 + t] = tanhf(s);                   // g in ft[64..127]
  }
  __syncthreads();
  float r = v1b[t];
  for (int k = 0; k < 128; ++k) r += ft[k] * v1W[t * 128 + k];
  r = r > 0.f ? r : 0.f;
  float p = r * v2W[t];
#pragma unroll
  for (int off = 16; off > 0; off >>= 1) p += __shfl_xor(p, off, 32);
  if ((t & 31) == 0) part[t >> 5] = p;
  __syncthreads();
  if (t == 0) out[0] = part[0] + part[1] + part[2] + part[3] + v2b[0];
}

// ---------- host ----------
static inline void zero4(float* p, long long n, hipStream_t stream) {
  const long long n4 = n >> 2;   // all zeroed sizes are multiples of 4
  zero4_f32<<<(unsigned)((n4 + 255) / 256), 256, 0, stream>>>((float4*)p, n4);
}

extern "C" void kernel_launch(void* const* d_in, const int* in_sizes, int n_in,
                              void* d_out, int out_size, void* d_ws, size_t ws_size,
                              hipStream_t stream) {
  const float* x          = (const float*)d_in[0];
  const int*   node_type  = (const int*)d_in[1];
  const int*   edge_index = (const int*)d_in[2];
  const float* edge_attr  = (const float*)d_in[3];
  const int*   comp_idx   = (const int*)d_in[4];
  const float* seq        = (const float*)d_in[5];
  const float* gWih = (const float*)d_in[6];
  const float* gWhh = (const float*)d_in[7];
  const float* gbih = (const float*)d_in[8];
  const float* gbhh = (const float*)d_in[9];
  const float* temb = (const float*)d_in[10];
  const float* linW = (const float*)d_in[11];
  const float* linb = (const float*)d_in[12];
  const float* g1Wl = (const float*)d_in[13];
  const float* g1Wr = (const float*)d_in[14];
  const float* g1We = (const float*)d_in[15];
  const float* g1att = (const float*)d_in[16];
  const float* g1b  = (const float*)d_in[17];
  const float* g2Wl = (const float*)d_in[18];
  const float* g2Wr = (const float*)d_in[19];
  const float* g2We = (const float*)d_in[20];
  const float* g2att = (const float*)d_in[21];
  const float* g2b  = (const float*)d_in[22];
  const float* lingW = (const float*)d_in[23];
  const float* lingb = (const float*)d_in[24];
  const float* v1W  = (const float*)d_in[25];
  const float* v1b  = (const float*)d_in[26];
  const float* v2W  = (const float*)d_in[27];
  const float* v2b  = (const float*)d_in[28];

  float* ws = (float*)d_ws;
  float*    Ht      = ws;                       // 10000*32
  float*    infeat  = Ht + 320000;              // 50000*72
  float*    hA      = infeat + 3600000;         // 50000*64
  float*    hB      = hA + 3200000;             // 50000*64
  float*    xl      = hB + 3200000;             // 50000*128
  float*    xr      = xl + 6400000;             // 50000*128
  float*    deg     = xr + 6400000;             // 50000
  float*    ealoop  = deg + 50000;              // 50000*8
  float*    score   = ealoop + 400000;          // 850000*2
  unsigned* smax    = (unsigned*)(score + 1700000); // 50000*2
  float*    denom   = (float*)smax + 100000;    // 50000*2
  float*    aggr    = denom + 100000;           // 50000*128
  float*    red     = aggr + 6400000;           // 128

  // 1) GRU over component sequences
  gru_kernel<<<(M_COMP + 7) / 8, 256, 0, stream>>>(seq, gWih, gWhh, gbih, gbhh, Ht);

  // 2) node input features: [x | scatter(Ht) | type_emb]
  fill_infeat_kernel<<<(int)(((long long)N_NODES * IN_FEAT + 255) / 256), 256, 0, stream>>>(
      x, node_type, temb, infeat);
  scatter_ht_kernel<<<(M_COMP * TEMP_HID + 255) / 256, 256, 0, stream>>>(comp_idx, Ht, infeat);

  // 3) h0 = relu(infeat @ lin_in_W^T + b)   (WMMA, K=72)
  gemm_wt<IN_FEAT, 1, true><<<3125, 128, 0, stream>>>(infeat, linW, linb, hA, N_NODES, HID);

  // 4) self-loop edge_attr = mean incoming edge_attr (layer-independent)
  zero4(deg, N_NODES, stream);
  zero4(ealoop, N_NODES * 8, stream);
  deg_easum_kernel<<<(N_EDGES + 255) / 256, 256, 0, stream>>>(edge_index, edge_attr, deg, ealoop);
  ealoop_div_kernel<<<(N_NODES * 8 + 255) / 256, 256, 0, stream>>>(deg, ealoop);

  // 5) GAT layer 1 (elu)  hA -> hB
  gemm_wt<64, 0, false><<<6250, 128, 0, stream>>>(hA, g1Wl, nullptr, xl, N_NODES, 128);
  gemm_wt<64, 0, false><<<6250, 128, 0, stream>>>(hA, g1Wr, nullptr, xr, N_NODES, 128);
  zero4((float*)smax, 2 * N_NODES, stream);
  zero4(denom, 2 * N_NODES, stream);
  zero4(aggr, (long long)N_NODES * 128, stream);
  edge_score_kernel<<<N_EDGES / 8, 256, 0, stream>>>(edge_index, edge_attr, xl, xr, g1We, g1att,
                                                     score, smax);
  loop_score_kernel<<<N_NODES / 8, 256, 0, stream>>>(ealoop, xl, xr, g1We, g1att, score, smax);
  expdenom_kernel<<<(E2 + 255) / 256, 256, 0, stream>>>(edge_index, score, smax, denom);
  aggregate_kernel<<<(E2 + 7) / 8, 256, 0, stream>>>(edge_index, score, denom, xl, aggr);
  finalize_kernel<<<(N_NODES * HID + 255) / 256, 256, 0, stream>>>(aggr, g1b, hB, 1);

  // 6) GAT layer 2 (no act)  hB -> hA  (H)
  gemm_wt<64, 0, false><<<6250, 128, 0, stream>>>(hB, g2Wl, nullptr, xl, N_NODES, 128);
  gemm_wt<64, 0, false><<<6250, 128, 0, stream>>>(hB, g2Wr, nullptr, xr, N_NODES, 128);
  zero4((float*)smax, 2 * N_NODES, stream);
  zero4(denom, 2 * N_NODES, stream);
  zero4(aggr, (long long)N_NODES * 128, stream);
  edge_score_kernel<<<N_EDGES / 8, 256, 0, stream>>>(edge_index, edge_attr, xl, xr, g2We, g2att,
                                                     score, smax);
  loop_score_kernel<<<N_NODES / 8, 256, 0, stream>>>(ealoop, xl, xr, g2We, g2att, score, smax);
  expdenom_kernel<<<(E2 + 255) / 256, 256, 0, stream>>>(edge_index, score, smax, denom);
  aggregate_kernel<<<(E2 + 7) / 8, 256, 0, stream>>>(edge_index, score, denom, xl, aggr);
  finalize_kernel<<<(N_NODES * HID + 255) / 256, 256, 0, stream>>>(aggr, g2b, hA, 0);

  // 7) reductions + value head
  zero4(red, 128, stream);
  sumH_kernel<<<(N_NODES + 127) / 128, 64, 0, stream>>>(hA, red);
  sumHc_kernel<<<(M_COMP + 127) / 128, 64, 0, stream>>>(hA, comp_idx, red);
  final_kernel<<<1, 128, 0, stream>>>(red, lingW, lingb, v1W, v1b, v2W, v2b, (float*)d_out);
}